// GraphConvolution_43026982371666
// MI455X (gfx1250) — compile-verified
//
#include <hip/hip_runtime.h>

typedef __attribute__((ext_vector_type(2))) float v2f;
typedef __attribute__((ext_vector_type(8))) float v8f;

#define NN     2048   // nodes
#define FIN    512
#define FOUT   512
#define FEDGE  16

// ---------------------------------------------------------------------------
// GEMM C[M,N] = A[M,K] @ B[K,N] with V_WMMA_F32_16X16X4_F32 (wave32).
// Optional fused epilogue: C = relu(A@B + E) when E != nullptr.
// Block: 256 threads = 8 waves; wave tile 32x32 (2x2 WMMA tiles);
// block tile 64(M) x 128(N). All dims divide evenly for this problem.
// ---------------------------------------------------------------------------
__global__ __launch_bounds__(256) void wmma_gemm_f32_kernel(
    const float* __restrict__ A, const float* __restrict__ B,
    const float* __restrict__ E, float* __restrict__ C,
    int M, int Ncols, int K)
{
    const int lane = threadIdx.x & 31;
    const int wave = threadIdx.x >> 5;

    const int wm = blockIdx.y * 64  + (wave & 1)  * 32;   // wave M origin
    const int wn = blockIdx.x * 128 + (wave >> 1) * 32;   // wave N origin

    const int ml = lane & 15;            // m (A) / n (B) within 16-tile
    const int kh = (lane >> 4) * 2;      // lane-half K offset: 0 or 2

    // A fragment pointers: per lane, contiguous float2 at (row, k+kh)
    const float* a0p = A + (size_t)(wm      + ml) * K + kh;
    const float* a1p = A + (size_t)(wm + 16 + ml) * K + kh;
    // B fragment pointers: per lane, B[kh][col], B[kh+1][col]
    const float* b0p = B + (size_t)kh * Ncols + wn + ml;
    const float* b1p = b0p + 16;
    const size_t bstep = (size_t)4 * Ncols;

    v8f c00 = {}, c01 = {}, c10 = {}, c11 = {};

    #pragma unroll 4
    for (int k = 0; k < K; k += 4) {
        v2f a0 = *(const v2f*)a0p;  a0p += 4;
        v2f a1 = *(const v2f*)a1p;  a1p += 4;
        v2f b0, b1;
        b0.x = b0p[0]; b0.y = b0p[Ncols];
        b1.x = b1p[0]; b1.y = b1p[Ncols];
        b0p += bstep;  b1p += bstep;

        c00 = __builtin_amdgcn_wmma_f32_16x16x4_f32(false, a0, false, b0, (short)0, c00, false, false);
        c01 = __builtin_amdgcn_wmma_f32_16x16x4_f32(false, a0, false, b1, (short)0, c01, false, false);
        c10 = __builtin_amdgcn_wmma_f32_16x16x4_f32(false, a1, false, b0, (short)0, c10, false, false);
        c11 = __builtin_amdgcn_wmma_f32_16x16x4_f32(false, a1, false, b1, (short)0, c11, false, false);
    }

    // C/D layout: VGPR r -> row = r + 8*(lane>>4), col = lane&15
    const int rbase = (lane >> 4) * 8;
    const int colb  = lane & 15;
    auto store_tile = [&](v8f c, int m0, int n0) {
        const int col = n0 + colb;
        #pragma unroll
        for (int r = 0; r < 8; ++r) {
            const int row = m0 + rbase + r;
            float v = c[r];
            if (E) {
                v += E[(size_t)row * Ncols + col];
                v = v > 0.0f ? v : 0.0f;
            }
            C[(size_t)row * Ncols + col] = v;
        }
    };
    store_tile(c00, wm,      wn);
    store_tile(c01, wm,      wn + 16);
    store_tile(c10, wm + 16, wn);
    store_tile(c11, wm + 16, wn + 16);
}

// ---------------------------------------------------------------------------
// Per-row i:  agg[f] = sum_j A[i,j] * Z[i,j,f]          (streams Z once, NT)
// then        E[i,c] = bias[c] + sum_f agg[f] * EK[f,c]
// One block of 256 threads per row. Thread t -> f = t&15, j-group = t>>4.
// Z accesses are perfectly coalesced (1KB contiguous per block iteration).
// Non-temporal loads keep the 256MB Z stream from evicting A/XW in L2.
// ---------------------------------------------------------------------------
__global__ __launch_bounds__(256) void edge_agg_kernel(
    const float* __restrict__ A, const float* __restrict__ Z,
    const float* __restrict__ EK, const float* __restrict__ bias,
    float* __restrict__ E)
{
    const int i  = blockIdx.x;
    const int t  = threadIdx.x;
    const int f  = t & 15;
    const int jg = t >> 4;

    __shared__ float red[256];

    const float* Arow = A + (size_t)i * NN;
    const float* Zrow = Z + (size_t)i * NN * FEDGE;

    float acc = 0.0f;
    #pragma unroll 4
    for (int j = jg; j < NN; j += 16)
        acc = fmaf(Arow[j], __builtin_nontemporal_load(Zrow + j * FEDGE + f), acc);

    red[t] = acc;
    __syncthreads();
    // reduce over the 16 j-groups (stride-16 layout keeps f contiguous)
    for (int s = 128; s >= 16; s >>= 1) {
        if (t < s) red[t] += red[t + s];
        __syncthreads();
    }
    // red[0..15] now holds agg[i, 0..15]

    for (int c = t; c < FOUT; c += 256) {
        float y = bias[c];
        #pragma unroll
        for (int ff = 0; ff < FEDGE; ++ff)
            y = fmaf(red[ff], EK[ff * FOUT + c], y);
        E[(size_t)i * FOUT + c] = y;
    }
}

extern "C" void kernel_launch(void* const* d_in, const int* in_sizes, int n_in,
                              void* d_out, int out_size, void* d_ws, size_t ws_size,
                              hipStream_t stream) {
    const float* X    = (const float*)d_in[0];  // [2048, 512]
    const float* A    = (const float*)d_in[1];  // [2048, 2048]
    const float* Z    = (const float*)d_in[2];  // [2048, 2048, 16]
    const float* W    = (const float*)d_in[3];  // [512, 512]
    const float* EK   = (const float*)d_in[4];  // [16, 512]
    const float* bias = (const float*)d_in[5];  // [512]
    float* out = (float*)d_out;                 // [2048, 512]

    float* XW = (float*)d_ws;                       // 2048*512 f32 = 4MB
    float* E  = XW + (size_t)NN * FOUT;             // 2048*512 f32 = 4MB

    // 1) XW = X @ W
    dim3 g1(FOUT / 128, NN / 64);
    wmma_gemm_f32_kernel<<<g1, 256, 0, stream>>>(X, W, nullptr, XW, NN, FOUT, FIN);

    // 2) E = agg @ edge_kernel + bias   (streams Z; dominant memory cost)
    edge_agg_kernel<<<NN, 256, 0, stream>>>(A, Z, EK, bias, E);

    // 3) out = relu(A @ XW + E)
    dim3 g2(FOUT / 128, NN / 64);
    wmma_gemm_f32_kernel<<<g2, 256, 0, stream>>>(A, XW, E, out, NN, FOUT, NN);
}